// snowball_layer_73280732004594
// MI455X (gfx1250) — compile-verified
//
#include <hip/hip_runtime.h>
#include <hip/hip_bf16.h>

// ---------------------------------------------------------------------------
// GCN layer: out = segment_sum( adj_vals[e] * (X W + b)[src[e]], dst[e] )
//   X: [50000,256] f32, W: [256,128] f32, b: [128] f32
//   h = X W + b  -> d_ws (25.6 MB)
//   out[i,:] += vals[e] * h[src[e],:] for dst[e]==i   (fp32 L2 atomics)
// ---------------------------------------------------------------------------

#define N_NODES 50000
#define N_EDGES 800000
#define IN_F    256
#define OUT_F   128

typedef float v2f __attribute__((ext_vector_type(2)));
typedef float v8f __attribute__((ext_vector_type(8)));

// ---------------------------------------------------------------------------
// Kernel 1: h = X*W + b via V_WMMA_F32_16X16X4_F32.
// Block = 256 threads = 8 waves. Block owns rows [16*bid, 16*bid+16);
// wave w owns cols [16*w, 16*w+16). K loop: 64 steps of K=4.
//
// fp32 WMMA lane layout (ISA 7.12.2):
//   A 16x4 : lanes 0-15 -> M=lane, V0=K0,V1=K1 ; lanes 16-31 -> M=lane-16, V0=K2,V1=K3
//   B 4x16 : lanes 0-15 -> N=lane, V0=K0,V1=K1 ; lanes 16-31 -> N=lane-16, V0=K2,V1=K3
//   C/D 16x16: VGPR r, lanes 0-15 -> M=r,N=lane ; lanes 16-31 -> M=r+8,N=lane-16
// ---------------------------------------------------------------------------
__global__ __launch_bounds__(256)
void gcn_gemm_wmma(const float* __restrict__ X,
                   const float* __restrict__ W,
                   const float* __restrict__ bias,
                   float* __restrict__ H)
{
    const int lane  = threadIdx.x & 31;
    const int wave  = threadIdx.x >> 5;          // 0..7 -> N tile
    const int mBase = blockIdx.x * 16;
    const int half  = lane >> 4;                 // 0 or 1
    const int l15   = lane & 15;
    const int kOff  = half * 2;                  // K sub-offset for this lane half

    const int row = mBase + l15;                 // A row for this lane
    const int col = wave * 16 + l15;             // B/C col for this lane

    const float* __restrict__ aPtr = X + (size_t)row * IN_F + kOff;
    const float* __restrict__ bPtr = W + (size_t)kOff * OUT_F + col;

    v8f acc = {};
    #pragma unroll 8
    for (int k = 0; k < IN_F; k += 4) {
        v2f a;
        a.x = aPtr[k];                           // K = k + kOff      (b64 load)
        a.y = aPtr[k + 1];                       // K = k + kOff + 1
        v2f b;
        b.x = bPtr[(size_t)k * OUT_F];           // W[k+kOff,   col]
        b.y = bPtr[(size_t)(k + 1) * OUT_F];     // W[k+kOff+1, col]
        acc = __builtin_amdgcn_wmma_f32_16x16x4_f32(
                  /*neg_a=*/false, a, /*neg_b=*/false, b,
                  /*c_mod=*/(short)0, acc, /*reuse_a=*/false, /*reuse_b=*/false);
    }

    const float bv   = bias[col];
    const int   mOff = mBase + half * 8;
    #pragma unroll
    for (int r = 0; r < 8; ++r) {
        H[(size_t)(mOff + r) * OUT_F + col] = acc[r] + bv;
    }
}

// ---------------------------------------------------------------------------
// Kernel 2: zero the output accumulator (float4 stores).
// ---------------------------------------------------------------------------
__global__ __launch_bounds__(256)
void gcn_zero_out(float4* __restrict__ out, int n4)
{
    int i = blockIdx.x * blockDim.x + threadIdx.x;
    if (i < n4) out[i] = make_float4(0.f, 0.f, 0.f, 0.f);
}

// ---------------------------------------------------------------------------
// Kernel 3: edge scatter. One wave per edge; lane l covers cols [4l,4l+4).
// h row (512 B) loads as b128; accumulation via global_atomic_add_f32,
// resolved in L2 (h and out both L2-resident: 2 x 25.6 MB << 192 MB).
// ---------------------------------------------------------------------------
__global__ __launch_bounds__(256)
void gcn_edge_scatter(const float* __restrict__ H,
                      const float* __restrict__ vals,
                      const int*   __restrict__ src,
                      const int*   __restrict__ dst,
                      float* __restrict__ out)
{
    const int lane = threadIdx.x & 31;
    const int wave = threadIdx.x >> 5;
    const int e    = blockIdx.x * 8 + wave;
    if (e >= N_EDGES) return;

    const int   s = src[e];
    const int   d = dst[e];
    const float v = vals[e];

    const float4 hv = ((const float4*)(H + (size_t)s * OUT_F))[lane];
    float* op = out + (size_t)d * OUT_F + lane * 4;

    unsafeAtomicAdd(op + 0, hv.x * v);
    unsafeAtomicAdd(op + 1, hv.y * v);
    unsafeAtomicAdd(op + 2, hv.z * v);
    unsafeAtomicAdd(op + 3, hv.w * v);
}

// ---------------------------------------------------------------------------
extern "C" void kernel_launch(void* const* d_in, const int* in_sizes, int n_in,
                              void* d_out, int out_size, void* d_ws, size_t ws_size,
                              hipStream_t stream)
{
    const float* X    = (const float*)d_in[0];   // [50000,256]
    const float* W    = (const float*)d_in[1];   // [256,128]
    const float* bias = (const float*)d_in[2];   // [128]
    const float* vals = (const float*)d_in[3];   // [800000]
    const int*   src  = (const int*)  d_in[4];   // [800000]
    const int*   dst  = (const int*)  d_in[5];   // [800000]
    float*       out  = (float*)d_out;           // [50000,128]
    float*       H    = (float*)d_ws;            // scratch: 50000*128*4 = 25.6 MB

    // 1) dense: h = X W + b   (3125 blocks x 8 waves, 16x16 tile per wave)
    gcn_gemm_wmma<<<N_NODES / 16, 256, 0, stream>>>(X, W, bias, H);

    // 2) out = 0
    const int n4 = (N_NODES * OUT_F) / 4;
    gcn_zero_out<<<(n4 + 255) / 256, 256, 0, stream>>>((float4*)out, n4);

    // 3) scatter-add over edges (8 edges per 256-thread block)
    gcn_edge_scatter<<<(N_EDGES + 7) / 8, 256, 0, stream>>>(H, vals, src, dst, out);
}